// SADRenderer_20847771255053
// MI455X (gfx1250) — compile-verified
//
#include <hip/hip_runtime.h>

#define INV_SCALE_SQ 0.25f

// Native 4-float vector for nontemporal builtins (HIP's float4 is a struct
// and is rejected by __builtin_nontemporal_store).
typedef float vf4 __attribute__((ext_vector_type(4)));

// ---------------------------------------------------------------------------
// Pass 1: repack sites [N,5] (20B rows, 4B-aligned) into gather-friendly
// layouts: pos as float2 (8B-aligned b64 gathers), col as padded float4
// (16B-aligned b128 gathers). Sites total 1.25MB -> stays hot in L2.
// ---------------------------------------------------------------------------
__global__ __launch_bounds__(256) void sites_repack_kernel(
    const float* __restrict__ sites,
    float2* __restrict__ pos,
    float4* __restrict__ col,
    int n) {
  int i = blockIdx.x * 256 + threadIdx.x;
  if (i >= n) return;
  const float* s = sites + 5ull * (unsigned)i;
  pos[i] = make_float2(s[0], s[1]);
  col[i] = make_float4(s[2], s[3], s[4], 0.0f);
}

// ---------------------------------------------------------------------------
// Pass 2: soft-Voronoi render. 4 pixels per thread, 1024 pixels per block.
//  - candidate index maps read as int4 (global_load_b128, coalesced)
//  - site gathers hit L2 (b64 + b128 in the PACKED path)
//  - sigmoid via v_exp_f32
//  - 12 floats/thread staged through LDS, then written as 3 coalesced
//    b128 non-temporal stores (output is write-once; don't pollute L2)
//  - global_prefetch of the next block's candidate tiles
// ---------------------------------------------------------------------------
template <bool PACKED>
__global__ __launch_bounds__(256) void voronoi_render_kernel(
    const float* __restrict__ sites,       // used when !PACKED
    const float2* __restrict__ spos,       // used when PACKED
    const float4* __restrict__ scol,       // used when PACKED
    const int4* __restrict__ cand0,
    const int4* __restrict__ cand1,
    const int* __restrict__ pW,
    float* __restrict__ out,
    int npix) {                            // npix % 4 == 0 (H*W)
  __shared__ float lds[1024 * 3];          // 12 KB

  const int tid = threadIdx.x;
  const int blockPix = blockIdx.x * 1024;
  const int t4 = blockPix / 4 + tid;       // int4 index into cand maps
  const unsigned W = (unsigned)*pW;

  // Speculative prefetch of next block's candidate tiles (gfx1250
  // global_prefetch_b8): hides HBM latency for the streaming inputs.
  if (blockPix + 2048 < npix) {
    __builtin_prefetch((const void*)(cand0 + t4 + 256), 0, 1);
    __builtin_prefetch((const void*)(cand1 + t4 + 256), 0, 1);
  }

  const unsigned p0 = (unsigned)(blockPix + 4 * tid);  // first pixel of thread
  if ((int)p0 < npix) {
    const int4 c0 = cand0[t4];
    const int4 c1 = cand1[t4];
    const int i0[4] = {c0.x, c0.y, c0.z, c0.w};
    const int i1[4] = {c1.x, c1.y, c1.z, c1.w};

    unsigned y = p0 / W;           // one u32 div per thread
    unsigned x = p0 - y * W;

    float acc[12];
#pragma unroll
    for (int k = 0; k < 4; ++k) {
      const float fx = (float)x + 0.5f;
      const float fy = (float)y + 0.5f;

      float sx0, sy0, r0, g0, b0, sx1, sy1, r1, g1, b1;
      if (PACKED) {
        const float2 P0 = spos[i0[k]];
        const float2 P1 = spos[i1[k]];
        const float4 C0 = scol[i0[k]];
        const float4 C1 = scol[i1[k]];
        sx0 = P0.x; sy0 = P0.y; r0 = C0.x; g0 = C0.y; b0 = C0.z;
        sx1 = P1.x; sy1 = P1.y; r1 = C1.x; g1 = C1.y; b1 = C1.z;
      } else {
        const float* s0 = sites + 5ull * (unsigned)i0[k];
        const float* s1 = sites + 5ull * (unsigned)i1[k];
        sx0 = s0[0]; sy0 = s0[1]; r0 = s0[2]; g0 = s0[3]; b0 = s0[4];
        sx1 = s1[0]; sy1 = s1[1]; r1 = s1[2]; g1 = s1[3]; b1 = s1[4];
      }

      const float dx0 = fx - sx0, dy0 = fy - sy0;
      const float dx1 = fx - sx1, dy1 = fy - sy1;
      const float d0 = dx0 * dx0 + dy0 * dy0;
      const float d1 = dx1 * dx1 + dy1 * dy1;
      const float z = (d1 - d0) * INV_SCALE_SQ;
      const float w = 1.0f / (1.0f + __expf(-z));   // v_exp_f32
      const float u = 1.0f - w;

      acc[3 * k + 0] = w * r0 + u * r1;
      acc[3 * k + 1] = w * g0 + u * g1;
      acc[3 * k + 2] = w * b0 + u * b1;

      if (++x == W) { x = 0; ++y; }      // pixels are row-major consecutive
    }

    // Stage to LDS (vectorizes to ds_store_b128 x3).
#pragma unroll
    for (int j = 0; j < 12; ++j) lds[12 * tid + j] = acc[j];
  }

  __syncthreads();

  // Re-read transposed so each wave's store instruction is a contiguous
  // 32-lane b128 burst: block writes 768 vf4s = 3 stores/thread.
  const vf4* lds4 = (const vf4*)lds;
  vf4* out4 = (vf4*)out;
  const long long base4 = (long long)blockIdx.x * 768;
  const long long tot4 = ((long long)npix * 3) >> 2;
#pragma unroll
  for (int j = 0; j < 3; ++j) {
    const long long o = base4 + 256 * j + tid;
    if (o < tot4) {
      __builtin_nontemporal_store(lds4[256 * j + tid], &out4[o]);
    }
  }
}

// ---------------------------------------------------------------------------
// Host launcher (graph-capture safe: only kernel launches on `stream`).
// Inputs: [0]=sites f32 [N*5], [1]=cand0 i32 [H*W], [2]=cand1 i32 [H*W],
//         [3]=width i32 [1], [4]=height i32 [1].  Output: f32 [H*W*3].
// ---------------------------------------------------------------------------
extern "C" void kernel_launch(void* const* d_in, const int* in_sizes, int n_in,
                              void* d_out, int out_size, void* d_ws, size_t ws_size,
                              hipStream_t stream) {
  (void)n_in; (void)out_size;
  const float* sites = (const float*)d_in[0];
  const int4* cand0 = (const int4*)d_in[1];
  const int4* cand1 = (const int4*)d_in[2];
  const int* pW = (const int*)d_in[3];
  float* out = (float*)d_out;

  const int nsite = in_sizes[0] / 5;
  const int npix = in_sizes[1];
  const int nblocks = (npix + 1023) / 1024;

  const size_t posBytes = (size_t)nsite * sizeof(float2);
  const size_t posBytesAligned = (posBytes + 255) & ~(size_t)255;
  const size_t colBytes = (size_t)nsite * sizeof(float4);
  const bool packed = (d_ws != nullptr) && (ws_size >= posBytesAligned + colBytes);

  if (packed) {
    float2* pos = (float2*)d_ws;
    float4* col = (float4*)((char*)d_ws + posBytesAligned);
    sites_repack_kernel<<<(nsite + 255) / 256, 256, 0, stream>>>(
        sites, pos, col, nsite);
    voronoi_render_kernel<true><<<nblocks, 256, 0, stream>>>(
        sites, pos, col, cand0, cand1, pW, out, npix);
  } else {
    voronoi_render_kernel<false><<<nblocks, 256, 0, stream>>>(
        sites, nullptr, nullptr, cand0, cand1, pW, out, npix);
  }
}